// HebbianPlasticityLayer_17282948399130
// MI455X (gfx1250) — compile-verified
//
#include <hip/hip_runtime.h>
#include <hip/hip_bf16.h>

// ---------------------------------------------------------------------------
// HebbianPlasticityLayer on MI455X (gfx1250), fp32 WMMA path.
// Phase 1: PP[N x 48] = x[N x 2048] @ [W_pre;W_post]^T   (reads x once, 256MB)
// Phase 2: modulation[N x 2048] = PP_pre[N x 24] @ Qt    (writes 256MB)
// where Qt[h][d] = sum_e effW[h,e] * W_out[d,e], effW from trace/hebbian rule.
// hebbian_state computed analytically from pre_mean @ effW.
// ---------------------------------------------------------------------------

typedef float v2f __attribute__((ext_vector_type(2)));
typedef float v8f __attribute__((ext_vector_type(8)));

#define DDIM   2048
#define HDIM   24
#define HC     48          // W_pre rows + W_post rows stacked
#define TDIM   4096
#define BDIM   8
#define NROWS  (BDIM * TDIM)   // 32768
#define KT     32              // K-chunk staged in LDS (phase 1)
#define RPB    128             // rows per block (phase 1) = 8 waves * 16
#define LPAD   33              // LDS row stride (bank-conflict padding)
#define DCHUNK 256             // d-columns per wave (phase 2)

// ---------------- Phase 1: PP = x @ Wc^T via V_WMMA_F32_16X16X4_F32 --------
__global__ __launch_bounds__(256) void hebb_phase1(
    const float* __restrict__ x, const float* __restrict__ Wpre,
    const float* __restrict__ Wpost, float* __restrict__ PP) {
  __shared__ float xtile[RPB][LPAD];   // 128 x 32 (padded)
  __shared__ float wtile[HC][LPAD];    // 48 x 32  (padded)

  const int tid  = threadIdx.x;
  const int wave = tid >> 5;
  const int lane = tid & 31;
  const int l16  = lane & 15;
  const int khalf = (lane >> 4) * 2;       // A/B fragment: lanes 16..31 hold K+2,K+3
  const int mbase = (lane >> 4) * 8;       // C fragment: lanes 16..31 hold M=8..15
  const int rowbase = blockIdx.x * RPB;
  const int arow = wave << 4;              // this wave's 16-row slice of the block

  v8f acc0 = {}, acc1 = {}, acc2 = {};     // 3 N-tiles of 16 cover HC=48

  const int ldr = tid >> 1;                // cooperative-load row (2 thr/row)
  const int ldc = (tid & 1) * 16;          // 16 floats each

  for (int k0 = 0; k0 < DDIM; k0 += KT) {
    // ---- stage x tile: 128 rows x 32 cols, coalesced float4 loads ----
    {
      const float* src = x + (size_t)(rowbase + ldr) * DDIM + k0 + ldc;
      #pragma unroll
      for (int q = 0; q < 4; ++q) {
        float4 v = *(const float4*)(src + q * 4);
        xtile[ldr][ldc + q * 4 + 0] = v.x;
        xtile[ldr][ldc + q * 4 + 1] = v.y;
        xtile[ldr][ldc + q * 4 + 2] = v.z;
        xtile[ldr][ldc + q * 4 + 3] = v.w;
      }
      if (k0 + KT < DDIM)  // gfx1250 global_prefetch_b8 for next chunk
        __builtin_prefetch(src + KT, 0, 1);
    }
    // ---- stage weight tile: 48 x 32 (pre rows 0..23, post rows 24..47) ----
    for (int i = tid; i < HC * (KT / 4); i += 256) {   // 384 float4s
      int h  = i >> 3;
      int c4 = (i & 7) * 4;
      const float* wsrc = (h < HDIM) ? (Wpre + (size_t)h * DDIM + k0 + c4)
                                     : (Wpost + (size_t)(h - HDIM) * DDIM + k0 + c4);
      float4 v = *(const float4*)wsrc;
      wtile[h][c4 + 0] = v.x; wtile[h][c4 + 1] = v.y;
      wtile[h][c4 + 2] = v.z; wtile[h][c4 + 3] = v.w;
    }
    __syncthreads();

    // ---- 8 K-steps of 4, 3 N-tiles each: V_WMMA_F32_16X16X4_F32 ----
    #pragma unroll
    for (int kk = 0; kk < KT; kk += 4) {
      v2f a, b;
      a.x = xtile[arow + l16][kk + khalf];
      a.y = xtile[arow + l16][kk + khalf + 1];

      b.x = wtile[l16][kk + khalf];
      b.y = wtile[l16][kk + khalf + 1];
      acc0 = __builtin_amdgcn_wmma_f32_16x16x4_f32(false, a, false, b,
                                                   (short)0, acc0, false, false);
      b.x = wtile[16 + l16][kk + khalf];
      b.y = wtile[16 + l16][kk + khalf + 1];
      acc1 = __builtin_amdgcn_wmma_f32_16x16x4_f32(false, a, false, b,
                                                   (short)0, acc1, false, false);
      b.x = wtile[32 + l16][kk + khalf];
      b.y = wtile[32 + l16][kk + khalf + 1];
      acc2 = __builtin_amdgcn_wmma_f32_16x16x4_f32(false, a, false, b,
                                                   (short)0, acc2, false, false);
    }
    __syncthreads();
  }

  // ---- write PP tile [16 x 48]: C layout -> lane=N, vgpr=M%8 ----
  const int row0 = rowbase + arow;
  #pragma unroll
  for (int r = 0; r < 8; ++r) {
    size_t base = (size_t)(row0 + mbase + r) * HC + l16;
    PP[base +  0] = acc0[r];
    PP[base + 16] = acc1[r];
    PP[base + 32] = acc2[r];
  }
}

// ---------------- means over T per batch: means[b][0..47] ------------------
__global__ void hebb_means(const float* __restrict__ PP, float* __restrict__ means) {
  int b = blockIdx.x;       // 0..7
  int h = threadIdx.x;      // 64 threads, 48 active
  if (h < HC) {
    float s = 0.f;
    const float* p = PP + (size_t)b * TDIM * HC + h;
    for (int t = 0; t < TDIM; ++t) s += p[(size_t)t * HC];
    means[b * HC + h] = s * (1.0f / (float)TDIM);
  }
}

// ------------- effW = W_plastic + (trace*DECAY + hebbian*LR); h_state ------
__global__ __launch_bounds__(576) void hebb_effw(
    const float* __restrict__ means, const float* __restrict__ Wplastic,
    const float* __restrict__ trace, float* __restrict__ effW,
    float* __restrict__ hs_out) {
  __shared__ float eff[HDIM * HDIM];
  int tid = threadIdx.x;
  if (tid < HDIM * HDIM) {
    int i = tid / HDIM, j = tid % HDIM;
    float hu = 0.f;
    #pragma unroll
    for (int b = 0; b < BDIM; ++b)
      hu += means[b * HC + i] * means[b * HC + HDIM + j];
    hu *= (1.0f / (float)BDIM);
    float e = Wplastic[tid] + (trace[tid] * 0.999f + hu * 0.01f);
    eff[tid]  = e;
    effW[tid] = e;
  }
  __syncthreads();
  if (tid < BDIM * HDIM) {   // hebbian_state[b][e] = sum_h pre_mean[b,h]*effW[h,e]
    int b = tid / HDIM, e = tid % HDIM;
    float s = 0.f;
    #pragma unroll
    for (int h = 0; h < HDIM; ++h) s += means[b * HC + h] * eff[h * HDIM + e];
    hs_out[tid] = s;
  }
}

// ------------- Qt[h][d] = sum_e effW[h,e] * W_out[d,e]  (h-major) ----------
__global__ __launch_bounds__(256) void hebb_qt(
    const float* __restrict__ effW, const float* __restrict__ Wout,
    float* __restrict__ Qt) {
  int idx = blockIdx.x * 256 + threadIdx.x;   // 24*2048 = 49152
  int h = idx >> 11;
  int d = idx & 2047;
  float s = 0.f;
  #pragma unroll
  for (int e = 0; e < HDIM; ++e) s += effW[h * HDIM + e] * Wout[(size_t)d * HDIM + e];
  Qt[(size_t)h * DDIM + d] = s;
}

// ------------- Phase 2: modulation = PP_pre @ Qt  (K = 24) -----------------
__global__ __launch_bounds__(128) void hebb_phase2(
    const float* __restrict__ PP, const float* __restrict__ Qt,
    float* __restrict__ mod) {
  const int tid  = threadIdx.x;
  const int wave = tid >> 5;
  const int lane = tid & 31;
  const int l16  = lane & 15;
  const int khalf = (lane >> 4) * 2;
  const int mbase = (lane >> 4) * 8;

  const int bid    = blockIdx.x;            // 4096 blocks
  const int dchunk = bid & 7;               // 8 d-chunks of 256
  const int rt     = (bid >> 3) * 4 + wave; // row tile 0..2047
  const int row0   = rt * 16;
  const int d0     = dchunk * DCHUNK;

  // A fragments for K=24 (6 steps of 4), reused across all 16 N-tiles
  v2f afrag[6];
  #pragma unroll
  for (int kk = 0; kk < 6; ++kk) {
    const float* p = PP + (size_t)(row0 + l16) * HC + kk * 4 + khalf; // pre cols 0..23
    afrag[kk].x = p[0];
    afrag[kk].y = p[1];
  }

  #pragma unroll 4
  for (int nt = 0; nt < DCHUNK / 16; ++nt) {
    const int dn = d0 + nt * 16 + l16;
    v8f acc = {};
    #pragma unroll
    for (int kk = 0; kk < 6; ++kk) {
      int k = kk * 4 + khalf;
      v2f b;
      b.x = Qt[(size_t)k * DDIM + dn];        // coalesced across lanes
      b.y = Qt[(size_t)(k + 1) * DDIM + dn];
      acc = __builtin_amdgcn_wmma_f32_16x16x4_f32(false, afrag[kk], false, b,
                                                  (short)0, acc, false, false);
    }
    #pragma unroll
    for (int r = 0; r < 8; ++r)
      mod[(size_t)(row0 + mbase + r) * DDIM + dn] = acc[r];
  }
}

// ---------------------------------------------------------------------------
extern "C" void kernel_launch(void* const* d_in, const int* in_sizes, int n_in,
                              void* d_out, int out_size, void* d_ws, size_t ws_size,
                              hipStream_t stream) {
  const float* x        = (const float*)d_in[0];  // [8,4096,2048]
  const float* Wpre     = (const float*)d_in[1];  // [24,2048]
  const float* Wpost    = (const float*)d_in[2];  // [24,2048]
  const float* Wplastic = (const float*)d_in[3];  // [24,24]
  const float* Wout     = (const float*)d_in[4];  // [2048,24]
  const float* trace    = (const float*)d_in[5];  // [24,24]

  float* mod = (float*)d_out;                              // 8*4096*2048 floats
  float* hs  = (float*)d_out + (size_t)NROWS * DDIM;       // + 192 floats

  float* ws    = (float*)d_ws;
  float* PP    = ws;                                       // 32768*48
  float* means = PP + (size_t)NROWS * HC;                  // 8*48
  float* effW  = means + BDIM * HC;                        // 24*24
  float* Qt    = effW + HDIM * HDIM;                       // 24*2048

  hipLaunchKernelGGL(hebb_phase1, dim3(NROWS / RPB), dim3(256), 0, stream,
                     x, Wpre, Wpost, PP);
  hipLaunchKernelGGL(hebb_means, dim3(BDIM), dim3(64), 0, stream, PP, means);
  hipLaunchKernelGGL(hebb_effw, dim3(1), dim3(576), 0, stream,
                     means, Wplastic, trace, effW, hs);
  hipLaunchKernelGGL(hebb_qt, dim3((HDIM * DDIM) / 256), dim3(256), 0, stream,
                     effW, Wout, Qt);
  hipLaunchKernelGGL(hebb_phase2, dim3((NROWS / 16 / 4) * 8), dim3(128), 0, stream,
                     PP, Qt, mod);
}